// _TopK_40072044872262
// MI455X (gfx1250) — compile-verified
//
#include <hip/hip_runtime.h>
#include <hip/hip_bf16.h>
#include <stdint.h>

#define B_ 16
#define N_ 32768
#define D_ 128
#define K_ 1024

typedef float v2f __attribute__((ext_vector_type(2)));
typedef float v8f __attribute__((ext_vector_type(8)));

__device__ __forceinline__ unsigned key_of_float(float f) {
    unsigned b = __float_as_uint(f);
    return (b & 0x80000000u) ? ~b : (b | 0x80000000u);
}
__device__ __forceinline__ float float_of_key(unsigned u) {
    unsigned b = (u & 0x80000000u) ? (u & 0x7FFFFFFFu) : ~u;
    return __uint_as_float(b);
}

// ---------------- Kernel 0: normalize p -> ps = p / ||p|| ----------------
__global__ __launch_bounds__(128) void prep_p_kernel(const float* __restrict__ p,
                                                     float* __restrict__ ps) {
    __shared__ float red[128];
    int t = threadIdx.x;
    float v = p[t];
    red[t] = v * v;
    __syncthreads();
    for (int s = 64; s > 0; s >>= 1) {
        if (t < s) red[t] += red[t + s];
        __syncthreads();
    }
    float inv = 1.0f / sqrtf(red[0]);
    ps[t] = v * inv;
}

// ---------------- Kernel 1: streaming scores via WMMA f32 16x16x4 ----------------
// Wave handles 16 consecutive rows. A-matrix layout for V_WMMA_F32_16X16X4_F32:
//   lanes 0-15 : row M=lane,    v0=K0, v1=K1
//   lanes 16-31: row M=lane-16, v0=K2, v1=K3
// -> each lane does one global_load_b64 at x[row*128 + 4c + 2*(lane>>4)].
// B = normalized p chunk replicated across all 16 columns (LDS broadcast loads).
// After 32 chunks, acc[g] on lane 0 holds dots of rows 0..7, lane 16 rows 8..15.
__global__ __launch_bounds__(256) void score_kernel(const float* __restrict__ x,
                                                    const float* __restrict__ ps,
                                                    unsigned* __restrict__ keys) {
    __shared__ __align__(16) float lp[D_];
    int tid = threadIdx.x;
    if (tid < D_) lp[tid] = ps[tid];
    __syncthreads();

    int lane = tid & 31;
    int wave = tid >> 5;
    int row_base = blockIdx.x * 128 + wave * 16;
    int m = lane & 15;
    int koff = (lane >> 4) * 2;   // 0 for lanes 0-15, 2 for lanes 16-31

    const float* rowp = x + (size_t)(row_base + m) * D_ + koff;
    __builtin_prefetch(rowp, 0, 0);   // global_prefetch_b8, speculative

    v8f acc = {0.f, 0.f, 0.f, 0.f, 0.f, 0.f, 0.f, 0.f};
#pragma unroll 8
    for (int c = 0; c < 32; ++c) {
        v2f a  = *(const v2f*)(rowp + 4 * c);          // A 16x4 tile slice
        v2f bm = *(const v2f*)(&lp[4 * c + koff]);     // B 4x16, all columns = ps chunk
        acc = __builtin_amdgcn_wmma_f32_16x16x4_f32(false, a, false, bm,
                                                    (short)0, acc, false, false);
    }

    // lane 0 owns rows row_base+0..7 (acc[0..7]); lane 16 owns rows +8..15
    if (lane == 0 || lane == 16) {
        unsigned* o = keys + row_base + ((lane >> 1) & 8);
#pragma unroll
        for (int g = 0; g < 8; ++g) o[g] = key_of_float(acc[g]);
    }
}

// ---------------- Kernel 2: exact per-batch top-1024 via bitonic merge ----------------
__device__ __forceinline__ void bitonic_sort_1024(unsigned long long* base, int t, bool desc) {
    for (int k = 2; k <= 1024; k <<= 1) {
        for (int j = k >> 1; j > 0; j >>= 1) {
            int ixj = t ^ j;
            if (ixj > t) {
                bool asc_block = (((t & k) == 0) != desc);
                unsigned long long a = base[t], b = base[ixj];
                if ((a > b) == asc_block) { base[t] = b; base[ixj] = a; }
            }
            __syncthreads();
        }
    }
}

__global__ __launch_bounds__(1024) void topk_kernel(const unsigned* __restrict__ keys,
                                                    unsigned* __restrict__ oidx,
                                                    unsigned* __restrict__ okey) {
    __shared__ unsigned long long s[2048];
    const int t = threadIdx.x;
    const unsigned* kb = keys + (size_t)blockIdx.x * N_;

    // chunk 0 -> descending result R in s[0..1023]
    s[t] = ((unsigned long long)kb[t] << 32) |
           (unsigned long long)(0xFFFFFFFFu - (unsigned)t);
    __syncthreads();
    bitonic_sort_1024(s, t, /*desc=*/true);

    for (int c = 1; c < N_ / 1024; ++c) {
        unsigned n = (unsigned)(c * 1024 + t);
        s[1024 + t] = ((unsigned long long)kb[n] << 32) |
                      (unsigned long long)(0xFFFFFFFFu - n);
        __syncthreads();
        bitonic_sort_1024(s + 1024, t, /*desc=*/false);  // ascending
        // s[0..2047] = (desc || asc) is bitonic: full merge to descending
        for (int j = 1024; j > 0; j >>= 1) {
            int i = ((t & ~(j - 1)) << 1) | (t & (j - 1));
            unsigned long long a = s[i], b = s[i + j];
            if (a < b) { s[i] = b; s[i + j] = a; }
            __syncthreads();
        }
    }

    unsigned long long r = s[t];
    oidx[blockIdx.x * K_ + t] = 0xFFFFFFFFu - (unsigned)r;
    okey[blockIdx.x * K_ + t] = (unsigned)(r >> 32);
}

// ---------------- Kernel 3: gather + tanh scale ----------------
__global__ __launch_bounds__(256) void gather_kernel(const float* __restrict__ x,
                                                     const unsigned* __restrict__ oidx,
                                                     const unsigned* __restrict__ okey,
                                                     float* __restrict__ out) {
    int tid = threadIdx.x;
    int lane = tid & 31;
    int wave = tid >> 5;
    int row = blockIdx.x * 8 + wave;        // 0 .. B*K-1
    int b = row >> 10;                      // row / K_
    unsigned idx = oidx[row];
    float tv = tanhf(float_of_key(okey[row]));

    const float4* src = (const float4*)(x + ((size_t)b * N_ + (size_t)idx) * D_) + lane;
    float4 v = *src;
    float4 r;
    r.x = v.x * tv; r.y = v.y * tv; r.z = v.z * tv; r.w = v.w * tv;
    *((float4*)(out + (size_t)row * D_) + lane) = r;
}

extern "C" void kernel_launch(void* const* d_in, const int* in_sizes, int n_in,
                              void* d_out, int out_size, void* d_ws, size_t ws_size,
                              hipStream_t stream) {
    const float* x = (const float*)d_in[0];   // [B, N, D] f32
    const float* p = (const float*)d_in[1];   // [D, 1]   f32
    (void)in_sizes; (void)n_in; (void)out_size; (void)ws_size;
    float* out = (float*)d_out;               // [B, K, D] f32

    // workspace: ps (512 B) | topk indices (64 KB) | topk score-keys (64 KB)
    float* ps = (float*)d_ws;
    unsigned* oidx = (unsigned*)((char*)d_ws + 512);
    unsigned* okey = oidx + B_ * K_;
    // score keys (2 MB) staged in front of d_out (8 MB); consumed by topk
    // before gather_kernel overwrites d_out.
    unsigned* keys = (unsigned*)d_out;

    prep_p_kernel<<<1, 128, 0, stream>>>(p, ps);
    score_kernel<<<(B_ * N_) / 128, 256, 0, stream>>>(x, ps, keys);
    topk_kernel<<<B_, 1024, 0, stream>>>(keys, oidx, okey);
    gather_kernel<<<(B_ * K_) / 8, 256, 0, stream>>>(x, oidx, okey, out);
}